// KMGCLModel_46797963657296
// MI455X (gfx1250) — compile-verified
//
#include <hip/hip_runtime.h>
#include <math.h>

typedef __attribute__((ext_vector_type(2))) float v2f;
typedef __attribute__((ext_vector_type(4))) float v4f;
typedef __attribute__((ext_vector_type(8))) float v8f;

#define NN 1024   // N == B
#define DD 128
#define NEG_INF (-3.402823466e38f)

// ---------------------------------------------------------------------------
// 1) L2-normalize rows of node (1024 x 128) and graph (1024 x 128).
//    One wave per row; each lane owns 4 contiguous floats (float4).
// ---------------------------------------------------------------------------
__global__ void k_normalize(const float* __restrict__ node,
                            const float* __restrict__ graph,
                            float* __restrict__ ne, float* __restrict__ ge) {
    int wave = threadIdx.x >> 5;
    int lane = threadIdx.x & 31;
    int row  = blockIdx.x * 8 + wave;         // 0..2047
    const float* src; float* dst;
    if (row < NN) { src = node  + row * DD;        dst = ne + row * DD; }
    else          { src = graph + (row - NN) * DD; dst = ge + (row - NN) * DD; }
    v4f v = *(const v4f*)(src + lane * 4);
    float ss = v.x * v.x + v.y * v.y + v.z * v.z + v.w * v.w;
#pragma unroll
    for (int off = 16; off >= 1; off >>= 1) ss += __shfl_xor(ss, off, 32);
    float inv = 1.0f / fmaxf(sqrtf(ss), 1e-12f);
    v.x *= inv; v.y *= inv; v.z *= inv; v.w *= inv;
    *(v4f*)(dst + lane * 4) = v;
}

// ---------------------------------------------------------------------------
// 2) C = E @ E^T via V_WMMA_F32_16X16X4_F32.  E is [1024 x 128] fp32 row-major.
//    One wave per 16x16 output tile, 32 WMMA steps of K=4.
//    A (16x4 f32) layout: lanes 0-15 hold row M=l with K = k,k+1 in the two
//    VGPRs; lanes 16-31 hold K = k+2,k+3.  B (4x16) mirrors with N=l.
//    C/D: VGPR r -> (M=r, N=lane) lanes 0-15, (M=r+8, N=lane-16) lanes 16-31.
// ---------------------------------------------------------------------------
__global__ void k_gemm_self(const float* __restrict__ E, float* __restrict__ C) {
    int wave = threadIdx.x >> 5;
    int lane = threadIdx.x & 31;
    int half = lane >> 4;        // 0 or 1
    int l    = lane & 15;
    int tile = blockIdx.x * 8 + wave;   // 0..4095
    int tm = (tile >> 6) * 16;
    int tn = (tile & 63) * 16;
    const float* arow = E + (tm + l) * DD + 2 * half;
    const float* brow = E + (tn + l) * DD + 2 * half;
    v8f acc = {0.f, 0.f, 0.f, 0.f, 0.f, 0.f, 0.f, 0.f};
#pragma unroll
    for (int k = 0; k < DD; k += 4) {
        v2f a = *(const v2f*)(arow + k);   // 8B-aligned (even element offset)
        v2f b = *(const v2f*)(brow + k);
        acc = __builtin_amdgcn_wmma_f32_16x16x4_f32(
            /*neg_a=*/false, a, /*neg_b=*/false, b,
            /*c_mod=*/(short)0, acc, /*reuse_a=*/false, /*reuse_b=*/false);
    }
    float* crow = C + (size_t)(tm + 8 * half) * NN + tn + l;
#pragma unroll
    for (int r = 0; r < 8; ++r) crow[(size_t)r * NN] = acc[r];
}

// ---------------------------------------------------------------------------
// 3) Row/col softmax stats.  grid = (128, 4); one wave per row/col.
//    mode 0: node_logits rows -> per-row node CE contribution (uses diag)
//    mode 1: graph_logits rows -> (max, sumexp)
//    mode 2: graph_logits cols -> (max, sumexp)   (faithful transposed stats)
//    mode 3: graph_metric rows -> (max, sumexp)
// ---------------------------------------------------------------------------
__global__ void k_stats(const float* __restrict__ nodeL,
                        const float* __restrict__ graphL,
                        const float* __restrict__ metric,
                        float* __restrict__ nodeContrib,
                        float* __restrict__ grm, float* __restrict__ grs,
                        float* __restrict__ gcm, float* __restrict__ gcs,
                        float* __restrict__ mrm, float* __restrict__ mrs) {
    int wave = threadIdx.x >> 5;
    int lane = threadIdx.x & 31;
    int i    = blockIdx.x * 8 + wave;
    int mode = blockIdx.y;
    const float* base; size_t stride;
    if      (mode == 0) { base = nodeL  + (size_t)i * NN; stride = 1; }
    else if (mode == 1) { base = graphL + (size_t)i * NN; stride = 1; }
    else if (mode == 2) { base = graphL + i;              stride = NN; }
    else                { base = metric + (size_t)i * NN; stride = 1; }

    float m = NEG_INF;
    for (int j = lane; j < NN; j += 32) m = fmaxf(m, base[(size_t)j * stride]);
#pragma unroll
    for (int off = 16; off >= 1; off >>= 1) m = fmaxf(m, __shfl_xor(m, off, 32));
    float s = 0.f;
    for (int j = lane; j < NN; j += 32) s += expf(base[(size_t)j * stride] - m);
#pragma unroll
    for (int off = 16; off >= 1; off >>= 1) s += __shfl_xor(s, off, 32);

    if (lane == 0) {
        if (mode == 0) {
            float diag = nodeL[(size_t)i * NN + i];
            nodeContrib[i] = -(diag - m - logf(s));   // -log_softmax diag
        } else if (mode == 1) { grm[i] = m; grs[i] = s; }
        else if   (mode == 2) { gcm[i] = m; gcs[i] = s; }
        else                  { mrm[i] = m; mrs[i] = s; }
    }
}

// ---------------------------------------------------------------------------
// 4) Per-row graph-loss partials.  One wave per row i.
//    p1 = sum_j gm[i][j] * lp[i][j]
//    p2 = sum_j gm[j][i] * lpT[i][j],  lpT from column stats of graph_logits.
// ---------------------------------------------------------------------------
__global__ void k_graph_loss(const float* __restrict__ G,
                             const float* __restrict__ metric,
                             const float* __restrict__ grm, const float* __restrict__ grs,
                             const float* __restrict__ gcm, const float* __restrict__ gcs,
                             const float* __restrict__ mrm, const float* __restrict__ mrs,
                             float* __restrict__ g1, float* __restrict__ g2) {
    int wave = threadIdx.x >> 5;
    int lane = threadIdx.x & 31;
    int i    = blockIdx.x * 8 + wave;
    float mi  = grm[i], lsi = logf(grs[i]);
    float Mi  = gcm[i], lSi = logf(gcs[i]);
    float ri  = mrm[i], ti  = mrs[i];
    float p1 = 0.f, p2 = 0.f;
    for (int j = lane; j < NN; j += 32) {
        float gmij = expf(metric[(size_t)i * NN + j] - ri) / ti;
        p1 += gmij * (G[(size_t)i * NN + j] - mi - lsi);
        float gmji = expf(metric[(size_t)j * NN + i] - mrm[j]) / mrs[j];
        p2 += gmji * (G[(size_t)j * NN + i] - Mi - lSi);
    }
#pragma unroll
    for (int off = 16; off >= 1; off >>= 1) {
        p1 += __shfl_xor(p1, off, 32);
        p2 += __shfl_xor(p2, off, 32);
    }
    if (lane == 0) { g1[i] = p1; g2[i] = p2; }
}

// ---------------------------------------------------------------------------
// 5) Deterministic final reduction: 1 block of 1024 threads (32 waves).
// ---------------------------------------------------------------------------
__global__ void k_final(const float* __restrict__ nodeContrib,
                        const float* __restrict__ g1, const float* __restrict__ g2,
                        float* __restrict__ out) {
    __shared__ float sm[3][32];
    int t = threadIdx.x, lane = t & 31, wave = t >> 5;
    float a = nodeContrib[t], b = g1[t], c = g2[t];
#pragma unroll
    for (int off = 16; off >= 1; off >>= 1) {
        a += __shfl_xor(a, off, 32);
        b += __shfl_xor(b, off, 32);
        c += __shfl_xor(c, off, 32);
    }
    if (lane == 0) { sm[0][wave] = a; sm[1][wave] = b; sm[2][wave] = c; }
    __syncthreads();
    if (wave == 0) {
        a = sm[0][lane]; b = sm[1][lane]; c = sm[2][lane];
#pragma unroll
        for (int off = 16; off >= 1; off >>= 1) {
            a += __shfl_xor(a, off, 32);
            b += __shfl_xor(b, off, 32);
            c += __shfl_xor(c, off, 32);
        }
        if (lane == 0) {
            float node_loss  = a / (float)NN;
            float graph_loss = -(b + c) / (float)NN;
            out[0] = 0.5f * node_loss + 0.5f * graph_loss;
            out[1] = node_loss;
            out[2] = graph_loss;
        }
    }
}

// ---------------------------------------------------------------------------
// 6) Pairwise concat writer: 1.074 GB of streaming stores -> non-temporal.
//    Block = 256 threads handles 4 output rows of 256 floats; consecutive
//    lanes write consecutive addresses (coalesced B32 NT stores; base is only
//    4B-aligned inside d_out, so B32 is the correct width).
// ---------------------------------------------------------------------------
__global__ void k_pairwise(const float* __restrict__ ne, float* __restrict__ outp) {
    int t = threadIdx.x;
    size_t rowBase = (size_t)blockIdx.x * 4;
#pragma unroll
    for (int r = 0; r < 4; ++r) {
        size_t row = rowBase + r;             // row index in [0, N*N)
        int i = (int)(row >> 10);
        int j = (int)(row & 1023);
        float v = (t < DD) ? ne[(size_t)i * DD + t] : ne[(size_t)j * DD + (t - DD)];
        __builtin_nontemporal_store(v, outp + row * (2 * DD) + t);
    }
}

// ---------------------------------------------------------------------------
extern "C" void kernel_launch(void* const* d_in, const int* in_sizes, int n_in,
                              void* d_out, int out_size, void* d_ws, size_t ws_size,
                              hipStream_t stream) {
    const float* node   = (const float*)d_in[0];   // [1024,128]
    const float* graph  = (const float*)d_in[1];   // [1024,128]
    const float* metric = (const float*)d_in[2];   // [1024,1024]

    float* out          = (float*)d_out;
    float* graph_logits = out + 3;                        // [1024,1024]
    float* pairwise     = out + 3 + (size_t)NN * NN;      // [N*N, 256]
    // node_logits staged in the (last-written) pairwise region: 4 MB of 1 GB.
    float* node_logits  = pairwise;

    float* ws  = (float*)d_ws;
    float* ne  = ws;                        // 1024*128
    float* ge  = ws + (size_t)NN * DD;      // 1024*128
    float* sa  = ws + 2 * (size_t)NN * DD;  // small arrays, 9 * 1024 floats
    float* nodeContrib = sa + 0 * NN;
    float* grm = sa + 1 * NN; float* grs = sa + 2 * NN;
    float* gcm = sa + 3 * NN; float* gcs = sa + 4 * NN;
    float* mrm = sa + 5 * NN; float* mrs = sa + 6 * NN;
    float* g1  = sa + 7 * NN; float* g2  = sa + 8 * NN;

    // 1) normalize (2048 rows, wave per row, 8 waves/block)
    k_normalize<<<256, 256, 0, stream>>>(node, graph, ne, ge);

    // 2) WMMA GEMMs: 4096 tiles, 8 waves/block -> 512 blocks each
    k_gemm_self<<<512, 256, 0, stream>>>(ne, node_logits);
    k_gemm_self<<<512, 256, 0, stream>>>(ge, graph_logits);

    // 3) softmax stats (4 modes x 1024 rows/cols)
    k_stats<<<dim3(128, 4), 256, 0, stream>>>(node_logits, graph_logits, metric,
                                              nodeContrib, grm, grs, gcm, gcs, mrm, mrs);

    // 4) per-row graph loss partials
    k_graph_loss<<<128, 256, 0, stream>>>(graph_logits, metric,
                                          grm, grs, gcm, gcs, mrm, mrs, g1, g2);

    // 5) scalars
    k_final<<<1, 1024, 0, stream>>>(nodeContrib, g1, g2, out);

    // 6) 1 GB pairwise stream (overwrites node_logits scratch) — launched last
    k_pairwise<<<(NN * NN) / 4, 256, 0, stream>>>(ne, pairwise);
}